// INRF_27711128994198
// MI455X (gfx1250) — compile-verified
//
#include <hip/hip_runtime.h>
#include <math.h>

// ---------------- problem constants ----------------
#define KS    21
#define FS    32.0f
#define H     48
#define W     48
#define C     3
#define NIMG  4
#define NPIX  (H * W * C)          // 6912 flattened pixels per image
#define PAD   (KS / 2)             // 10
#define TOT   (NIMG * NPIX)        // 27648

// pairwise stage tiling
#define WAVES_PER_BLOCK 8
#define TILES_I   (NPIX / 16)      // 432 sixteen-row tiles per image
#define BLOCKS_X  (TILES_I / WAVES_PER_BLOCK)  // 54
#define JCHUNKS   4
#define JCHUNK_LEN (NPIX / JCHUNKS)            // 1728 (divisible by 16)

typedef float v2f __attribute__((ext_vector_type(2)));
typedef float v8f __attribute__((ext_vector_type(8)));

// ---------------------------------------------------------------------------
// Branch-free tanh. Prefer the native gfx1250 V_TANH_F32; otherwise use a
// straight-line exp2/rcp formulation that never touches EXEC (required both
// for speed and because the surrounding WMMA stream wants full EXEC).
// ---------------------------------------------------------------------------
__device__ __forceinline__ float fast_tanh(float x) {
#if __has_builtin(__builtin_amdgcn_tanhf)
    return __builtin_amdgcn_tanhf(x);
#elif __has_builtin(__builtin_amdgcn_tanh_f32)
    return __builtin_amdgcn_tanh_f32(x);
#else
    // tanh(x) = sign(x) * (1 - e) / (1 + e),  e = exp(-2|x|) in (0,1]
    const float ax = __builtin_fabsf(x);
    const float e  = __builtin_amdgcn_exp2f(ax * -2.8853900817779268f); // -2*log2(e)
    const float t  = (1.0f - e) * __builtin_amdgcn_rcpf(1.0f + e);
    return __builtin_copysignf(t, x);
#endif
}

// ---------------------------------------------------------------------------
// Stage A: build the three energy-normalized Gaussian tap sets on device.
// Amp factor cancels under L2 normalization, so tap = exp(-r^2*gamma^2/2)/||.||2
// K layout: K[(set*3 + c)*441 + dy*21 + dx], set: 0=m, 1=w, 2=g
// ---------------------------------------------------------------------------
__global__ void inrf_make_kernels(const float* __restrict__ gm,
                                  const float* __restrict__ gw,
                                  const float* __restrict__ gg,
                                  float* __restrict__ K) {
    __shared__ float red[512];
    const int t = threadIdx.x;          // 512 threads
    const float step = (KS / FS) / (float)(KS - 1);
    const float mean = KS / (2.0f * FS);

    for (int set = 0; set < 3; ++set) {
        const float* g = (set == 0) ? gm : ((set == 1) ? gw : gg);
        for (int c = 0; c < C; ++c) {
            float val = 0.0f;
            if (t < KS * KS) {
                const int dy = t / KS, dx = t % KS;
                const float x = dx * step - mean;
                const float y = dy * step - mean;
                const float r2 = x * x + y * y;
                const float gamma = g[c];
                val = expf(-0.5f * r2 * gamma * gamma);
            }
            red[t] = val * val;
            __syncthreads();
            for (int s = 256; s > 0; s >>= 1) {
                if (t < s) red[t] += red[t + s];
                __syncthreads();
            }
            const float norm = rsqrtf(red[0]);
            if (t < KS * KS) K[(set * 3 + c) * (KS * KS) + t] = val * norm;
            __syncthreads();
        }
    }
}

// ---------------------------------------------------------------------------
// Stage B: fused depthwise convs with the M and G filters (shared neighborhood).
// ---------------------------------------------------------------------------
__global__ void inrf_conv_mg(const float* __restrict__ in,
                             const float* __restrict__ K,
                             float* __restrict__ first_term,
                             float* __restrict__ blurred) {
    const int idx = blockIdx.x * blockDim.x + threadIdx.x;
    if (idx >= TOT) return;
    const int c = idx % C;
    const int w = (idx / C) % W;
    const int h = (idx / (C * W)) % H;
    const int n = idx / NPIX;

    const float* km = K + (0 * 3 + c) * (KS * KS);
    const float* kg = K + (2 * 3 + c) * (KS * KS);
    const float* base = in + n * NPIX;

    float am = 0.0f, ag = 0.0f;
    for (int dy = 0; dy < KS; ++dy) {
        const int y = h + dy - PAD;
        if (y < 0 || y >= H) continue;
        for (int dx = 0; dx < KS; ++dx) {
            const int x = w + dx - PAD;
            if (x < 0 || x >= W) continue;
            const float v = base[(y * W + x) * C + c];
            am = fmaf(v, km[dy * KS + dx], am);
            ag = fmaf(v, kg[dy * KS + dx], ag);
        }
    }
    first_term[idx] = am;
    blurred[idx]    = ag;
}

// ---------------------------------------------------------------------------
// Stage C: pairwise tanh reduction via V_WMMA_F32_16X16X4_F32.
//   difference[n,i] = (1/6912) * sum_j tanh(x[n,i] - blurred[n,j])
// One wave owns a 16-row i-tile; j advances by 16 per iteration through FOUR
// independent WMMA accumulators (breaks the C-operand RAW chain, pipelines
// the matrix unit, and gives the VALU 8 independent tanh ops per WMMA).
// A-fragment built directly in the 16x4 f32 layout (lanes0-15: M=row,K=0..1 ;
// lanes16-31: same rows, K=2..3). B = all-ones (layout independent).
// grid: (54, JCHUNKS, NIMG), block: 256
// ---------------------------------------------------------------------------
__global__ void inrf_pairwise(const float* __restrict__ in,
                              const float* __restrict__ blurred,
                              float* __restrict__ diff) {
    __shared__ float sb[NPIX];                 // 27 KB: whole blurred image
    const int n  = blockIdx.z;
    const int jc = blockIdx.y;
    const int tid = threadIdx.x;

    for (int t = tid; t < NPIX; t += 256)
        sb[t] = blurred[n * NPIX + t];
    __syncthreads();

    const int wave = tid >> 5;
    const int lane = tid & 31;
    const int row  = lane & 15;
    const int k2   = lane >> 4;                // float2 index: lanes0-15 -> 0, 16-31 -> 1

    const int tile  = blockIdx.x * WAVES_PER_BLOCK + wave;
    const int ibase = tile * 16;
    const float xi = in[n * NPIX + ibase + row];

    const int j0 = jc * JCHUNK_LEN;
    const int j1 = j0 + JCHUNK_LEN;

    const float2* sb2 = reinterpret_cast<const float2*>(sb);

    v2f ones; ones.x = 1.0f; ones.y = 1.0f;
    v8f acc0 = {}, acc1 = {}, acc2 = {}, acc3 = {};

    for (int j = j0; j < j1; j += 16) {
        const int p = (j >> 1) + k2;           // float2 slot for this lane
        const float2 b0 = sb2[p + 0];
        const float2 b1 = sb2[p + 2];
        const float2 b2 = sb2[p + 4];
        const float2 b3 = sb2[p + 6];

        v2f a0, a1, a2, a3;
        a0.x = fast_tanh(xi - b0.x);  a0.y = fast_tanh(xi - b0.y);
        a1.x = fast_tanh(xi - b1.x);  a1.y = fast_tanh(xi - b1.y);
        a2.x = fast_tanh(xi - b2.x);  a2.y = fast_tanh(xi - b2.y);
        a3.x = fast_tanh(xi - b3.x);  a3.y = fast_tanh(xi - b3.y);

        acc0 = __builtin_amdgcn_wmma_f32_16x16x4_f32(
                   false, a0, false, ones, (short)0, acc0, false, false);
        acc1 = __builtin_amdgcn_wmma_f32_16x16x4_f32(
                   false, a1, false, ones, (short)0, acc1, false, false);
        acc2 = __builtin_amdgcn_wmma_f32_16x16x4_f32(
                   false, a2, false, ones, (short)0, acc2, false, false);
        acc3 = __builtin_amdgcn_wmma_f32_16x16x4_f32(
                   false, a3, false, ones, (short)0, acc3, false, false);
    }

    const v8f acc = (acc0 + acc1) + (acc2 + acc3);

    const float scale = 1.0f / (float)NPIX;
    // C layout: VGPR r holds M=r (lanes 0-15) and M=8+r (lanes 16-31)
    if (lane == 0) {
        #pragma unroll
        for (int r = 0; r < 8; ++r)
            atomicAdd(&diff[n * NPIX + ibase + r], acc[r] * scale);
    }
    if (lane == 16) {
        #pragma unroll
        for (int r = 0; r < 8; ++r)
            atomicAdd(&diff[n * NPIX + ibase + 8 + r], acc[r] * scale);
    }
}

// ---------------------------------------------------------------------------
// Stage D: out = first_term - conv(difference, K_w)
// ---------------------------------------------------------------------------
__global__ void inrf_conv_w_out(const float* __restrict__ diff,
                                const float* __restrict__ K,
                                const float* __restrict__ first_term,
                                float* __restrict__ out) {
    const int idx = blockIdx.x * blockDim.x + threadIdx.x;
    if (idx >= TOT) return;
    const int c = idx % C;
    const int w = (idx / C) % W;
    const int h = (idx / (C * W)) % H;
    const int n = idx / NPIX;

    const float* kw = K + (1 * 3 + c) * (KS * KS);
    const float* base = diff + n * NPIX;

    float aw = 0.0f;
    for (int dy = 0; dy < KS; ++dy) {
        const int y = h + dy - PAD;
        if (y < 0 || y >= H) continue;
        for (int dx = 0; dx < KS; ++dx) {
            const int x = w + dx - PAD;
            if (x < 0 || x >= W) continue;
            aw = fmaf(base[(y * W + x) * C + c], kw[dy * KS + dx], aw);
        }
    }
    out[idx] = first_term[idx] - aw;
}

// ---------------------------------------------------------------------------
extern "C" void kernel_launch(void* const* d_in, const int* in_sizes, int n_in,
                              void* d_out, int out_size, void* d_ws, size_t ws_size,
                              hipStream_t stream) {
    (void)in_sizes; (void)n_in; (void)out_size; (void)ws_size;

    const float* inputs = (const float*)d_in[0];
    const float* gm     = (const float*)d_in[1];
    const float* gw     = (const float*)d_in[2];
    const float* gg     = (const float*)d_in[3];
    float* out = (float*)d_out;

    float* ws    = (float*)d_ws;
    float* K     = ws;                 // 3*3*441 = 3969 floats (pad to 4096)
    float* first = ws + 4096;          // 27648
    float* blur  = first + TOT;        // 27648
    float* diff  = blur + TOT;         // 27648

    inrf_make_kernels<<<1, 512, 0, stream>>>(gm, gw, gg, K);

    inrf_conv_mg<<<(TOT + 255) / 256, 256, 0, stream>>>(inputs, K, first, blur);

    hipMemsetAsync(diff, 0, TOT * sizeof(float), stream);

    inrf_pairwise<<<dim3(BLOCKS_X, JCHUNKS, NIMG), 256, 0, stream>>>(inputs, blur, diff);

    inrf_conv_w_out<<<(TOT + 255) / 256, 256, 0, stream>>>(diff, K, first, out);
}